// GCN_Estimator_37503654429288
// MI455X (gfx1250) — compile-verified
//
#include <hip/hip_runtime.h>

typedef __attribute__((ext_vector_type(2))) float v2f;
typedef __attribute__((ext_vector_type(8))) float v8f;

#define NUM_U   40000
#define NUM_I   60000
#define DIM     128
#define NNODE   (NUM_U + NUM_I)
#define NNZE    1600000
#define BATCH   4096
#define HID1    512
#define HID2    256
#define FC1_IN  1024   // 8 * DIM

// ---------------- init: x = concat(user_emb, item_emb); acc = x ----------------
__global__ void init_x_acc(const float* __restrict__ ue, const float* __restrict__ ie,
                           float* __restrict__ x, float* __restrict__ acc) {
    long i = (long)blockIdx.x * blockDim.x + threadIdx.x;       // float4 index
    const long total = (long)NNODE * DIM / 4;
    if (i >= total) return;
    const long uelems = (long)NUM_U * DIM / 4;
    float4 v = (i < uelems) ? ((const float4*)ue)[i] : ((const float4*)ie)[i - uelems];
    ((float4*)x)[i]   = v;
    ((float4*)acc)[i] = v;
}

__global__ void zero_f4(float* __restrict__ p, long n4) {
    long i = (long)blockIdx.x * blockDim.x + threadIdx.x;
    if (i >= n4) return;
    ((float4*)p)[i] = make_float4(0.f, 0.f, 0.f, 0.f);
}

__global__ void acc_add_f4(float* __restrict__ acc, const float* __restrict__ y, long n4) {
    long i = (long)blockIdx.x * blockDim.x + threadIdx.x;
    if (i >= n4) return;
    float4 a = ((const float4*)acc)[i];
    float4 b = ((const float4*)y)[i];
    a.x += b.x; a.y += b.y; a.z += b.z; a.w += b.w;
    ((float4*)acc)[i] = a;
}

// ---------------- COO SpMM: y[row] += val * x[col]; one wave per nnz ----------------
// x (51.2MB) is L2-resident; lane owns 4 consecutive floats of the 128-wide row.
__global__ void spmm_coo(const int* __restrict__ row, const int* __restrict__ col,
                         const float* __restrict__ val,
                         const float* __restrict__ x, float* __restrict__ y) {
    long t = (long)blockIdx.x * blockDim.x + threadIdx.x;
    long e = t >> 5;                 // wave-uniform nonzero index
    if (e >= NNZE) return;
    int lane = (int)(t & 31);
    int r = row[e];
    int c = col[e];
    float v = val[e];
    float4 xv = ((const float4*)(x + (long)c * DIM))[lane];     // coalesced 512B gather
    float* yp = y + (long)r * DIM + lane * 4;
    __hip_atomic_fetch_add(yp + 0, v * xv.x, __ATOMIC_RELAXED, __HIP_MEMORY_SCOPE_AGENT);
    __hip_atomic_fetch_add(yp + 1, v * xv.y, __ATOMIC_RELAXED, __HIP_MEMORY_SCOPE_AGENT);
    __hip_atomic_fetch_add(yp + 2, v * xv.z, __ATOMIC_RELAXED, __HIP_MEMORY_SCOPE_AGENT);
    __hip_atomic_fetch_add(yp + 3, v * xv.w, __ATOMIC_RELAXED, __HIP_MEMORY_SCOPE_AGENT);
}

// ---------------- gather GCN rows (scaled by 1/(L+1)=0.25) into X[:,0:256] ----------------
__global__ void gather_gcn(const int* __restrict__ uid, const int* __restrict__ iid,
                           const float* __restrict__ acc, float* __restrict__ X) {
    long t = (long)blockIdx.x * blockDim.x + threadIdx.x;   // BATCH * 32 lanes
    if (t >= (long)BATCH * 32) return;
    int b = (int)(t >> 5);
    int q = (int)(t & 31);                                  // float4 slot within 128-wide row
    float4 u = ((const float4*)(acc + (long)uid[b] * DIM))[q];
    float4 it = ((const float4*)(acc + (long)(NUM_U + iid[b]) * DIM))[q];
    u.x *= 0.25f; u.y *= 0.25f; u.z *= 0.25f; u.w *= 0.25f;
    it.x *= 0.25f; it.y *= 0.25f; it.z *= 0.25f; it.w *= 0.25f;
    ((float4*)(X + (long)b * FC1_IN))[q] = u;
    ((float4*)(X + (long)b * FC1_IN + DIM))[q] = it;
}

// ---------------- fp32 WMMA GEMM: C[:, ccol:ccol+N] = relu?(A@B + bias) ----------------
// One wave computes a 16(M) x 64(N) strip: 4 independent 16x16 accumulators
// (4 interleaved v_wmma_f32_16x16x4_f32 chains hide the XDL C->C latency),
// A fragment loaded once per K-step and reused by all 4 tiles.
// A: [M,K] row-major (lda), B: [K,N] row-major (ldb), C row-major (ldc).
__global__ __launch_bounds__(128) void gemm_wmma_f32(
        const float* __restrict__ A, const float* __restrict__ B,
        const float* __restrict__ bias, float* __restrict__ C,
        int M, int N, int K, int lda, int ldb, int ldc, int ccol, int relu) {
    int groupsN = N >> 6;                       // strips of 64 columns
    int numGroups = (M >> 4) * groupsN;
    int g = blockIdx.x * blockDim.y + threadIdx.y;   // wave-uniform
    if (g >= numGroups) return;
    int gm = g / groupsN;
    int gn = g % groupsN;
    int lane = threadIdx.x;        // 0..31
    int mloc = lane & 15;          // M (for A) / N (for B/C) within a 16-tile
    int hi   = lane >> 4;          // half-wave: selects K+2..3 pair, and M+8 rows of C
    int khalf = hi * 2;
    int m0 = gm * 16;
    int n0 = gn * 64;

    const float* Arow = A + (long)(m0 + mloc) * lda + khalf;
    const float* Bp   = B + (long)khalf * ldb + n0 + mloc;   // tile t at Bp + t*16

    v8f acc0 = {}, acc1 = {}, acc2 = {}, acc3 = {};
    for (int k = 0; k < K; k += 4) {
        v2f a;
        a.x = Arow[0];
        a.y = Arow[1];
        Arow += 4;
        v2f b0, b1, b2, b3;
        b0.x = Bp[0];        b0.y = Bp[ldb];
        b1.x = Bp[16];       b1.y = Bp[ldb + 16];
        b2.x = Bp[32];       b2.y = Bp[ldb + 32];
        b3.x = Bp[48];       b3.y = Bp[ldb + 48];
        Bp += (long)4 * ldb;
        acc0 = __builtin_amdgcn_wmma_f32_16x16x4_f32(false, a, false, b0, (short)0, acc0, false, false);
        acc1 = __builtin_amdgcn_wmma_f32_16x16x4_f32(false, a, false, b1, (short)0, acc1, false, false);
        acc2 = __builtin_amdgcn_wmma_f32_16x16x4_f32(false, a, false, b2, (short)0, acc2, false, false);
        acc3 = __builtin_amdgcn_wmma_f32_16x16x4_f32(false, a, false, b3, (short)0, acc3, false, false);
    }

    float* Crow = C + (long)(m0 + hi * 8) * ldc + ccol + n0 + mloc;
    v8f accs[4] = {acc0, acc1, acc2, acc3};
#pragma unroll
    for (int t = 0; t < 4; ++t) {
        int n = n0 + t * 16 + mloc;
        float bv = bias ? bias[n] : 0.0f;
#pragma unroll
        for (int r2 = 0; r2 < 8; ++r2) {
            float vv = accs[t][r2] + bv;
            if (relu) vv = fmaxf(vv, 0.0f);
            Crow[(long)r2 * ldc + t * 16] = vv;
        }
    }
}

// ---------------- out layer: [B,256] @ [256,1] + b, wave per row ----------------
__global__ void out_layer(const float* __restrict__ Hin, const float* __restrict__ W,
                          const float* __restrict__ bptr, float* __restrict__ out) {
    long t = (long)blockIdx.x * blockDim.x + threadIdx.x;
    int r = (int)(t >> 5);
    int lane = (int)(t & 31);
    if (r >= BATCH) return;
    const float* h = Hin + (long)r * HID2;
    float s = 0.0f;
#pragma unroll
    for (int j = 0; j < HID2 / 32; ++j) s += h[lane + 32 * j] * W[lane + 32 * j];
#pragma unroll
    for (int off = 16; off > 0; off >>= 1) s += __shfl_xor(s, off, 32);
    if (lane == 0) out[r] = s + bptr[0];
}

extern "C" void kernel_launch(void* const* d_in, const int* in_sizes, int n_in,
                              void* d_out, int out_size, void* d_ws, size_t ws_size,
                              hipStream_t stream) {
    (void)in_sizes; (void)n_in; (void)out_size; (void)ws_size;
    const float* uf0 = (const float*)d_in[0];
    const float* uf1 = (const float*)d_in[1];
    const float* uf2 = (const float*)d_in[2];
    const float* if0 = (const float*)d_in[3];
    const float* if1 = (const float*)d_in[4];
    const float* if2 = (const float*)d_in[5];
    const int*   uid = (const int*)d_in[6];
    const int*   iid = (const int*)d_in[7];
    const int*   arow = (const int*)d_in[8];
    const int*   acol = (const int*)d_in[9];
    const float* aval = (const float*)d_in[10];
    const float* uemb = (const float*)d_in[11];
    const float* iemb = (const float*)d_in[12];
    const float* Wu0 = (const float*)d_in[13];
    const float* Wu1 = (const float*)d_in[14];
    const float* Wu2 = (const float*)d_in[15];
    const float* Wi0 = (const float*)d_in[16];
    const float* Wi1 = (const float*)d_in[17];
    const float* Wi2 = (const float*)d_in[18];
    const float* fc1W = (const float*)d_in[19];
    const float* fc1b = (const float*)d_in[20];
    const float* fc2W = (const float*)d_in[21];
    const float* fc2b = (const float*)d_in[22];
    const float* outW = (const float*)d_in[23];
    const float* outb = (const float*)d_in[24];
    float* out = (float*)d_out;

    // workspace carve-up
    float* xb   = (float*)d_ws;                                  // [N, D]
    float* yb   = xb   + (size_t)NNODE * DIM;                    // [N, D]
    float* accb = yb   + (size_t)NNODE * DIM;                    // [N, D]
    float* Xb   = accb + (size_t)NNODE * DIM;                    // [B, 1024]
    float* H1b  = Xb   + (size_t)BATCH * FC1_IN;                 // [B, 512]
    float* H2b  = H1b  + (size_t)BATCH * HID1;                   // [B, 256]

    const long n4 = (long)NNODE * DIM / 4;
    const int TB = 256;

    init_x_acc<<<(unsigned)((n4 + TB - 1) / TB), TB, 0, stream>>>(uemb, iemb, xb, accb);

    float* xcur = xb;
    float* ycur = yb;
    for (int l = 0; l < 3; ++l) {
        zero_f4<<<(unsigned)((n4 + TB - 1) / TB), TB, 0, stream>>>(ycur, n4);
        long sthreads = (long)NNZE * 32;
        spmm_coo<<<(unsigned)((sthreads + TB - 1) / TB), TB, 0, stream>>>(arow, acol, aval, xcur, ycur);
        acc_add_f4<<<(unsigned)((n4 + TB - 1) / TB), TB, 0, stream>>>(accb, ycur, n4);
        float* tmp = xcur; xcur = ycur; ycur = tmp;
    }

    gather_gcn<<<(BATCH * 32 + TB - 1) / TB, TB, 0, stream>>>(uid, iid, accb, Xb);

    auto launch_gemm = [&](const float* A, const float* Bm, const float* bias, float* C,
                           int M, int N, int K, int lda, int ldb, int ldc, int ccol, int relu) {
        int groups = (M / 16) * (N / 64);
        dim3 blk(32, 4);
        gemm_wmma_f32<<<(groups + 3) / 4, blk, 0, stream>>>(A, Bm, bias, C,
                                                            M, N, K, lda, ldb, ldc, ccol, relu);
    };

    // feature projections into X column slices (X layout: [u_gcn | i_gcn | emb_u x3 | emb_i x3])
    launch_gemm(uf0, Wu0, nullptr, Xb, BATCH, DIM, 64, 64, DIM, FC1_IN, 256, 0);
    launch_gemm(uf1, Wu1, nullptr, Xb, BATCH, DIM, 32, 32, DIM, FC1_IN, 384, 0);
    launch_gemm(uf2, Wu2, nullptr, Xb, BATCH, DIM, 16, 16, DIM, FC1_IN, 512, 0);
    launch_gemm(if0, Wi0, nullptr, Xb, BATCH, DIM, 64, 64, DIM, FC1_IN, 640, 0);
    launch_gemm(if1, Wi1, nullptr, Xb, BATCH, DIM, 32, 32, DIM, FC1_IN, 768, 0);
    launch_gemm(if2, Wi2, nullptr, Xb, BATCH, DIM, 16, 16, DIM, FC1_IN, 896, 0);

    // MLP
    launch_gemm(Xb,  fc1W, fc1b, H1b, BATCH, HID1, FC1_IN, FC1_IN, HID1, HID1, 0, 1);
    launch_gemm(H1b, fc2W, fc2b, H2b, BATCH, HID2, HID1,   HID1,   HID2, HID2, 0, 1);

    out_layer<<<(BATCH * 32 + TB - 1) / TB, TB, 0, stream>>>(H2b, outW, outb, out);
}